// Net_3702261809246
// MI455X (gfx1250) — compile-verified
//
#include <hip/hip_runtime.h>
#include <math.h>

#define N_NODES 50000
#define N_EDGES 1600000
#define E_TOT   (N_EDGES + N_NODES)
#define LEAKY   0.2f
#define BN_EPS  1e-5f
#define BN_BLOCKS 512
#define EPB 8

typedef __attribute__((ext_vector_type(2))) float v2f;
typedef __attribute__((ext_vector_type(8))) float v8f;

// ---- order-preserving float<->int encoding for atomicMax on signed int ----
__device__ __forceinline__ int f2ord(float f) {
    int i = __float_as_int(f);
    return (i >= 0) ? i : (i ^ 0x7fffffff);
}
__device__ __forceinline__ float ord2f(int k) {
    return __int_as_float((k >= 0) ? k : (k ^ 0x7fffffff));
}
__device__ __forceinline__ float lrelu(float v) { return v > 0.f ? v : LEAKY * v; }

// ---------------- BatchNorm statistics (deterministic 2-pass) ----------------
__global__ void bn_partial_k(const float* __restrict__ X, float* __restrict__ part) {
    int f = threadIdx.x;                     // 128 threads, one per feature
    float s = 0.f, q = 0.f;
    for (int r = blockIdx.x; r < N_NODES; r += BN_BLOCKS) {
        float v = X[r * 128 + f];
        s += v; q += v * v;
    }
    part[blockIdx.x * 256 + f]       = s;
    part[blockIdx.x * 256 + 128 + f] = q;
}

__global__ void bn_finalize_k(const float* __restrict__ part,
                              const float* __restrict__ bg, const float* __restrict__ bb,
                              float* __restrict__ sc, float* __restrict__ sh) {
    int f = threadIdx.x;                     // 128 threads
    float s = 0.f, q = 0.f;
    for (int b = 0; b < BN_BLOCKS; ++b) {
        s += part[b * 256 + f];
        q += part[b * 256 + 128 + f];
    }
    float mean  = s / (float)N_NODES;
    float var   = q / (float)N_NODES - mean * mean;
    float scale = bg[f] * rsqrtf(var + BN_EPS);
    sc[f] = scale;
    sh[f] = bb[f] - mean * scale;
}

// ------------- fused BN + GEMM via f32 WMMA: C = BN(A) @ B (+bias) -----------
// A:[N,128] B:[128,128] C:[N,128]. Block = 256 thr = 8 waves; block covers 16
// rows, each wave one 16-col strip. 32x v_wmma_f32_16x16x4_f32 per wave.
__global__ __launch_bounds__(256) void gemm_bn_wmma_k(
    const float* __restrict__ A, const float* __restrict__ sc, const float* __restrict__ sh,
    const float* __restrict__ B, const float* __restrict__ bias1, const float* __restrict__ bias2,
    float* __restrict__ C) {
    __shared__ float As[16 * 130];           // pad 130: bank-conflict-free, 8B aligned
    int row0 = blockIdx.x * 16;
    int tid  = threadIdx.x;
    for (int i = tid; i < 16 * 128; i += 256) {
        int r = i >> 7, k = i & 127;
        As[r * 130 + k] = A[(row0 + r) * 128 + k] * sc[k] + sh[k];   // fused BN
    }
    __syncthreads();

    int lane = tid & 31;
    int n0   = (tid >> 5) * 16;              // wave's column strip
    int l15  = lane & 15;                    // A row / B col within tile
    int kb   = (lane >> 4) << 1;             // lanes 16-31 hold K+2,K+3
    v8f acc = {};
    #pragma unroll
    for (int k0 = 0; k0 < 128; k0 += 4) {
        v2f a, b;
        a.x = As[l15 * 130 + k0 + kb];
        a.y = As[l15 * 130 + k0 + kb + 1];
        b.x = B[(k0 + kb)     * 128 + n0 + l15];
        b.y = B[(k0 + kb + 1) * 128 + n0 + l15];
        acc = __builtin_amdgcn_wmma_f32_16x16x4_f32(
                  false, a, false, b, (short)0, acc, false, false);
    }
    float bsum = 0.f;
    if (bias1) bsum += bias1[n0 + l15];
    if (bias2) bsum += bias2[n0 + l15];
    int mhi = (lane >> 4) * 8;
    #pragma unroll
    for (int j = 0; j < 8; ++j)
        C[(row0 + j + mhi) * 128 + n0 + l15] = acc[j] + bsum;
}

// --------- attention logits: es/ed[n,h] = sum_f hw[n,h,f] * a[h,f] ----------
__global__ void attn_proj_k(const float* __restrict__ hw, const float* __restrict__ a_s,
                            const float* __restrict__ a_d, int H, int F,
                            float* __restrict__ es, float* __restrict__ ed) {
    int i = blockIdx.x * blockDim.x + threadIdx.x;
    if (i >= N_NODES * H) return;
    int node = i / H, h = i % H;
    const float* row = hw + (size_t)node * H * F + h * F;
    float s = 0.f, d = 0.f;
    for (int f = 0; f < F; ++f) {
        float v = row[f];
        s += v * a_s[h * F + f];
        d += v * a_d[h * F + f];
    }
    es[i] = s; ed[i] = d;
}

__global__ void init_ms_k(int* __restrict__ m, float* __restrict__ s, int count) {
    int i = blockIdx.x * blockDim.x + threadIdx.x;
    if (i < count) { m[i] = f2ord(-INFINITY); s[i] = 0.f; }
}

__device__ __forceinline__ void edge_sd(long long e, const long long* ei, int& s_, int& d_) {
    if (e < N_EDGES) { s_ = (int)ei[e]; d_ = (int)ei[N_EDGES + e]; }
    else             { s_ = d_ = (int)(e - N_EDGES); }    // self-loops
}

// -------------------- segment max of leaky-relu logits ----------------------
__global__ void edge_max_k(const long long* __restrict__ ei,
                           const float* __restrict__ es, const float* __restrict__ ed,
                           int H, int* __restrict__ m) {
    long long e = (long long)blockIdx.x * blockDim.x + threadIdx.x;
    if (e >= E_TOT) return;
    int s_, d_; edge_sd(e, ei, s_, d_);
    for (int h = 0; h < H; ++h) {
        float v = lrelu(es[s_ * H + h] + ed[d_ * H + h]);
        atomicMax(&m[d_ * H + h], f2ord(v));
    }
}

// -------------------- softmax denominator per destination -------------------
__global__ void edge_sum_k(const long long* __restrict__ ei,
                           const float* __restrict__ es, const float* __restrict__ ed,
                           const int* __restrict__ m, int H, float* __restrict__ ssum) {
    long long e = (long long)blockIdx.x * blockDim.x + threadIdx.x;
    if (e >= E_TOT) return;
    int s_, d_; edge_sd(e, ei, s_, d_);
    for (int h = 0; h < H; ++h) {
        float v = lrelu(es[s_ * H + h] + ed[d_ * H + h]);
        atomicAdd(&ssum[d_ * H + h], expf(v - ord2f(m[d_ * H + h])));
    }
}

// ---------- weighted aggregation, layers 1/2 (C=128, H=2, F=64) -------------
// 128 threads = one output row per edge: 512B contiguous gather + row atomics.
__global__ __launch_bounds__(128) void edge_agg128_k(
    const long long* __restrict__ ei,
    const float* __restrict__ es, const float* __restrict__ ed,
    const int* __restrict__ m, const float* __restrict__ ssum,
    const float* __restrict__ hw, float* __restrict__ acc) {
    int c = threadIdx.x;
    int h = c >> 6;
    long long base = (long long)blockIdx.x * EPB;
    for (int i = 0; i < EPB; ++i) {
        long long e = base + i;
        if (e >= E_TOT) return;
        int s_, d_; edge_sd(e, ei, s_, d_);
        float v = lrelu(es[s_ * 2 + h] + ed[d_ * 2 + h]);
        float coef = expf(v - ord2f(m[d_ * 2 + h])) / ssum[d_ * 2 + h];
        atomicAdd(&acc[(size_t)d_ * 128 + c], coef * hw[(size_t)s_ * 128 + c]);
    }
}

// --------------------- layer 3 fused projection (C=2) -----------------------
__global__ void layer3_proj_k(const float* __restrict__ A,
                              const float* __restrict__ sc, const float* __restrict__ sh,
                              const float* __restrict__ W3, const float* __restrict__ lw3,
                              const float* __restrict__ lb3, const float* __restrict__ gb3,
                              const float* __restrict__ as3, const float* __restrict__ ad3,
                              float* __restrict__ hw3, float* __restrict__ acc3,
                              float* __restrict__ es3, float* __restrict__ ed3) {
    int node = blockIdx.x * blockDim.x + threadIdx.x;
    if (node >= N_NODES) return;
    float h0 = 0.f, h1 = 0.f, l0 = 0.f, l1 = 0.f;
    for (int k = 0; k < 128; ++k) {
        float v = A[(size_t)node * 128 + k] * sc[k] + sh[k];
        h0 += v * W3[k * 2 + 0];  h1 += v * W3[k * 2 + 1];
        l0 += v * lw3[k * 2 + 0]; l1 += v * lw3[k * 2 + 1];
    }
    hw3[node * 2 + 0] = h0;  hw3[node * 2 + 1] = h1;
    acc3[node * 2 + 0] = l0 + lb3[0] + gb3[0];
    acc3[node * 2 + 1] = l1 + lb3[1] + gb3[1];
    es3[node] = h0 * as3[0] + h1 * as3[1];
    ed3[node] = h0 * ad3[0] + h1 * ad3[1];
}

__global__ void edge_agg2_k(const long long* __restrict__ ei,
                            const float* __restrict__ es, const float* __restrict__ ed,
                            const int* __restrict__ m, const float* __restrict__ ssum,
                            const float* __restrict__ hw, float* __restrict__ acc) {
    long long e = (long long)blockIdx.x * blockDim.x + threadIdx.x;
    if (e >= E_TOT) return;
    int s_, d_; edge_sd(e, ei, s_, d_);
    float v = lrelu(es[s_] + ed[d_]);
    float coef = expf(v - ord2f(m[d_])) / ssum[d_];
    atomicAdd(&acc[d_ * 2 + 0], coef * hw[s_ * 2 + 0]);
    atomicAdd(&acc[d_ * 2 + 1], coef * hw[s_ * 2 + 1]);
}

__global__ void relu_k(const float* __restrict__ in, float* __restrict__ out, int count) {
    int i = blockIdx.x * blockDim.x + threadIdx.x;
    if (i < count) out[i] = fmaxf(in[i], 0.f);
}

// ----------------------------------------------------------------------------
extern "C" void kernel_launch(void* const* d_in, const int* in_sizes, int n_in,
                              void* d_out, int out_size, void* d_ws, size_t ws_size,
                              hipStream_t stream) {
    (void)in_sizes; (void)n_in; (void)out_size; (void)ws_size;
    const float*     x  = (const float*)d_in[0];
    const long long* ei = (const long long*)d_in[1];
    const float* W [3] = {(const float*)d_in[2],  (const float*)d_in[10], (const float*)d_in[18]};
    const float* AS[3] = {(const float*)d_in[3],  (const float*)d_in[11], (const float*)d_in[19]};
    const float* AD[3] = {(const float*)d_in[4],  (const float*)d_in[12], (const float*)d_in[20]};
    const float* GB[3] = {(const float*)d_in[5],  (const float*)d_in[13], (const float*)d_in[21]};
    const float* LW[3] = {(const float*)d_in[6],  (const float*)d_in[14], (const float*)d_in[22]};
    const float* LB[3] = {(const float*)d_in[7],  (const float*)d_in[15], (const float*)d_in[23]};
    const float* BG[3] = {(const float*)d_in[8],  (const float*)d_in[16], (const float*)d_in[24]};
    const float* BB[3] = {(const float*)d_in[9],  (const float*)d_in[17], (const float*)d_in[25]};

    float* ws = (float*)d_ws;
    size_t o = 0;
    float* A    = ws + o; o += (size_t)N_NODES * 128;   // layer output / next input
    float* HW   = ws + o; o += (size_t)N_NODES * 128;   // h @ W (also hw3 with stride 2)
    float* ACC  = ws + o; o += (size_t)N_NODES * 128;   // lin + bias, then += gat
    float* ES   = ws + o; o += (size_t)N_NODES * 2;
    float* ED   = ws + o; o += (size_t)N_NODES * 2;
    int*   M    = (int*)(ws + o); o += (size_t)N_NODES * 2;
    float* S    = ws + o; o += (size_t)N_NODES * 2;
    float* PART = ws + o; o += (size_t)BN_BLOCKS * 256;
    float* SC   = ws + o; o += 128;
    float* SH   = ws + o; o += 128;

    const int eg = (E_TOT + 255) / 256;
    const float* hin = x;

    // ---------------- layers 1 & 2 (H=2, F=64, C=128) ----------------
    for (int l = 0; l < 2; ++l) {
        bn_partial_k <<<BN_BLOCKS, 128, 0, stream>>>(hin, PART);
        bn_finalize_k<<<1, 128, 0, stream>>>(PART, BG[l], BB[l], SC, SH);
        gemm_bn_wmma_k<<<N_NODES / 16, 256, 0, stream>>>(hin, SC, SH, W[l],  nullptr, nullptr, HW);
        gemm_bn_wmma_k<<<N_NODES / 16, 256, 0, stream>>>(hin, SC, SH, LW[l], LB[l],  GB[l],   ACC);
        attn_proj_k<<<(N_NODES * 2 + 255) / 256, 256, 0, stream>>>(HW, AS[l], AD[l], 2, 64, ES, ED);
        init_ms_k  <<<(N_NODES * 2 + 255) / 256, 256, 0, stream>>>(M, S, N_NODES * 2);
        edge_max_k <<<eg, 256, 0, stream>>>(ei, ES, ED, 2, M);
        edge_sum_k <<<eg, 256, 0, stream>>>(ei, ES, ED, M, 2, S);
        edge_agg128_k<<<(E_TOT + EPB - 1) / EPB, 128, 0, stream>>>(ei, ES, ED, M, S, HW, ACC);
        relu_k<<<(N_NODES * 128 + 255) / 256, 256, 0, stream>>>(ACC, A, N_NODES * 128);
        hin = A;
    }

    // ---------------- layer 3 (H=1, F=2, C=2) ----------------
    bn_partial_k <<<BN_BLOCKS, 128, 0, stream>>>(hin, PART);
    bn_finalize_k<<<1, 128, 0, stream>>>(PART, BG[2], BB[2], SC, SH);
    layer3_proj_k<<<(N_NODES + 255) / 256, 256, 0, stream>>>(
        hin, SC, SH, W[2], LW[2], LB[2], GB[2], AS[2], AD[2], HW, ACC, ES, ED);
    init_ms_k <<<(N_NODES + 255) / 256, 256, 0, stream>>>(M, S, N_NODES);
    edge_max_k<<<eg, 256, 0, stream>>>(ei, ES, ED, 1, M);
    edge_sum_k<<<eg, 256, 0, stream>>>(ei, ES, ED, M, 1, S);
    edge_agg2_k<<<eg, 256, 0, stream>>>(ei, ES, ED, M, S, HW, ACC);
    relu_k<<<(N_NODES * 2 + 255) / 256, 256, 0, stream>>>(ACC, (float*)d_out, N_NODES * 2);
}